// BasketEmbedding_22514218565933
// MI455X (gfx1250) — compile-verified
//
#include <hip/hip_runtime.h>

// BasketEmbedding pooled gather for MI455X (gfx1250).
// Inputs (setup_inputs order):
//   d_in[0] table       f32 [100001, 128]
//   d_in[1] item_ids    i32 [1024, 50, 20]
//   d_in[2] basket_lens i32 [1024, 50]
//   d_in[3] seq_lens    i32 [1024]
// Output: f32 [1024, 50, 128]

#define B_DIM   1024
#define S_DIM   50
#define M_DIM   20
#define H_DIM   128
#define N_ITEMS 100000

typedef float v4f __attribute__((ext_vector_type(4)));  // native vector for NT stores

constexpr int WAVES_PER_BLOCK = 8;
constexpr int THREADS_PER_BLOCK = WAVES_PER_BLOCK * 32;

__global__ __launch_bounds__(THREADS_PER_BLOCK) void
BasketEmbedding_gfx1250_kernel(const float* __restrict__ table,
                               const int*   __restrict__ item_ids,
                               const int*   __restrict__ basket_lens,
                               const int*   __restrict__ seq_lens,
                               float*       __restrict__ out)
{
    // Per-wave staging slab: M_DIM rows x H_DIM f32 = 10 KB per wave.
    __shared__ float smem[WAVES_PER_BLOCK][M_DIM * H_DIM];

    const int lane = threadIdx.x & 31;          // wave32 on CDNA5
    const int widx = threadIdx.x >> 5;
    const int wave = blockIdx.x * WAVES_PER_BLOCK + widx;  // basket index (b*S + s)
    if (wave >= B_DIM * S_DIM) return;

    const int b = wave / S_DIM;
    const int s = wave - b * S_DIM;

    v4f* orow = reinterpret_cast<v4f*>(out + (size_t)wave * H_DIM) + lane;

    // Sequence padding: constant vector of value N_ITEMS. Uniform branch per wave.
    const int slen = seq_lens[b];
    if (s >= slen) {
        const float pv = (float)N_ITEMS;
        const v4f pad = {pv, pv, pv, pv};
        __builtin_nontemporal_store(pad, orow);
        return;
    }

    const int len = basket_lens[wave];          // uniform, in [1, 20]
    const int* ids = item_ids + (size_t)wave * M_DIM;

    // One coalesced fetch of the 20 wave-uniform ids.
    int myid = (lane < M_DIM) ? ids[lane] : 0;

    // This lane's 16-byte column chunk within every gathered row.
    const unsigned lane_off = (unsigned)(lane * 16);

    // Workgroup-relative LDS byte offset of this wave's slab (flat addr low 32 bits).
    const unsigned lds_base =
        (unsigned)(size_t)(&smem[widx][0]) + lane_off;

    // ---- Fire the whole gather asynchronously into LDS (ASYNCcnt path) ----
    // GVS addressing: SGPR pair = table base, VGPR = 32-bit byte offset
    // (id*512 + lane*16 <= ~51.2 MB, fits easily in 32 bits).
    for (int m = 0; m < len; ++m) {
        const int id = __builtin_amdgcn_readlane(myid, m);     // uniform -> SGPR
        const unsigned goff  = (unsigned)id * (H_DIM * 4) + lane_off;
        const unsigned laddr = lds_base + (unsigned)m * (H_DIM * 4);
        asm volatile("global_load_async_to_lds_b128 %0, %1, %2"
                     :
                     : "v"(laddr), "v"(goff), "s"(table)
                     : "memory");
    }

    // Drain: all async copies for this wave landed in its own slab; no barrier needed.
    asm volatile("s_wait_asynccnt 0x0" ::: "memory");

    // ---- Accumulate from LDS (ds_load_b128 per row) ----
    v4f acc = {0.f, 0.f, 0.f, 0.f};
    const v4f* lrow =
        reinterpret_cast<const v4f*>(&smem[widx][lane * 4]);
    for (int m = 0; m < len; ++m) {
        acc += lrow[m * (H_DIM / 4)];
    }

    const float inv = 1.0f / (float)len;        // mean pooling
    acc *= inv;
    __builtin_nontemporal_store(acc, orow);
}

extern "C" void kernel_launch(void* const* d_in, const int* in_sizes, int n_in,
                              void* d_out, int out_size, void* d_ws, size_t ws_size,
                              hipStream_t stream)
{
    (void)in_sizes; (void)n_in; (void)out_size; (void)d_ws; (void)ws_size;

    const float* table       = (const float*)d_in[0];
    const int*   item_ids    = (const int*)  d_in[1];
    const int*   basket_lens = (const int*)  d_in[2];
    const int*   seq_lens    = (const int*)  d_in[3];
    float*       out         = (float*)d_out;

    const int total_waves = B_DIM * S_DIM;                     // 51200 baskets
    const int blocks = (total_waves + WAVES_PER_BLOCK - 1) / WAVES_PER_BLOCK;  // 6400

    BasketEmbedding_gfx1250_kernel<<<blocks, THREADS_PER_BLOCK, 0, stream>>>(
        table, item_ids, basket_lens, seq_lens, out);
}